// Psudo_MultiView_Keyp_Loss_72670846649076
// MI455X (gfx1250) — compile-verified
//
#include <hip/hip_runtime.h>

typedef float v2f __attribute__((ext_vector_type(2)));
typedef float v8f __attribute__((ext_vector_type(8)));

#define J_N 26
#define WAVES_PER_BLOCK 8

struct F3 { float x, y, z; };

__device__ __forceinline__ F3 f3add(F3 a, F3 b) { return {a.x + b.x, a.y + b.y, a.z + b.z}; }
__device__ __forceinline__ F3 f3sub(F3 a, F3 b) { return {a.x - b.x, a.y - b.y, a.z - b.z}; }
// Rodrigues about x-axis: R90 v = (x, -z, y); RM90 v = (x, z, -y); R180 v = (x, -y, -z)
__device__ __forceinline__ F3 rot90(F3 v)  { return { v.x, -v.z,  v.y}; }
__device__ __forceinline__ F3 rotm90(F3 v) { return { v.x,  v.z, -v.y}; }
__device__ __forceinline__ F3 rot180(F3 v) { return { v.x, -v.y, -v.z}; }
__device__ __forceinline__ F3 f3neg(F3 v)  { return {-v.x, -v.y, -v.z}; }

// f*xy/z projection of p and g, squared diff of the two 2D points
__device__ __forceinline__ float projd(F3 p, F3 g, float f) {
    float izp = 1.0f / p.z, izg = 1.0f / g.z;
    float dx = f * (p.x * izp) - f * (g.x * izg);
    float dy = f * (p.y * izp) - f * (g.y * izg);
    return dx * dx + dy * dy;
}

// One wave32 per batch: lanes 0..25 handle joints. Coalesced loads
// (gt rows are 16B-aligned float4), butterfly-shuffle reductions for the
// centroid means and for the per-batch partial loss. One f32 partial / block.
__global__ __launch_bounds__(WAVES_PER_BLOCK * 32)
void mvk_loss_partial(const float* __restrict__ pred_joints,   // [B,J,3]
                      const float* __restrict__ pred_trans,    // [B,3]
                      const float* __restrict__ gt_joints,     // [B,J,4]
                      const float* __restrict__ gt_trans,      // [B,3]
                      const float* __restrict__ focal,         // [B]
                      float* __restrict__ partials)            // [gridDim.x]
{
    __shared__ float lds[WAVES_PER_BLOCK];
    const int wave = threadIdx.x >> 5;
    const int lane = threadIdx.x & 31;
    const int b = blockIdx.x * WAVES_PER_BLOCK + wave;

    const float tpx = pred_trans[b * 3 + 0], tpy = pred_trans[b * 3 + 1], tpz = pred_trans[b * 3 + 2];
    const float tgx = gt_trans[b * 3 + 0],  tgy = gt_trans[b * 3 + 1],  tgz = gt_trans[b * 3 + 2];
    const float f = focal[b];

    F3 p = {0.f, 0.f, 1.f}, g = {0.f, 0.f, 1.f};
    float gx = 0.f, gy = 0.f, gz = 0.f;
    const bool act = (lane < J_N);
    if (act) {
        const float* pj = pred_joints + ((size_t)b * J_N + lane) * 3;
        p = {pj[0] + tpx, pj[1] + tpy, pj[2] + tpz};
        const float4 gj = *(const float4*)(gt_joints + ((size_t)b * J_N + lane) * 4);
        g = {gj.x + tgx, gj.y + tgy, gj.z + tgz};
        gx = g.x; gy = g.y; gz = g.z;
    }
    // wave sums of gt components -> centroid means
    #pragma unroll
    for (int m = 16; m >= 1; m >>= 1) {
        gx += __shfl_xor(gx, m, 32);
        gy += __shfl_xor(gy, m, 32);
        gz += __shfl_xor(gz, m, 32);
    }
    const float inv_j = 1.0f / (float)J_N;
    const float mx = gx * inv_j, my = gy * inv_j, mz = gz * inv_j;
    const F3 c1 = {mx, my, 0.f};
    // c2 = zero_z(mean(g1)) = zero_z(R90(mean(gt)-c1)+c1) = (mx, my - mz, 0)
    const F3 c2 = {mx, my - mz, 0.f};

    float s = 0.f;
    if (act) {
        F3 q = f3sub(p, c1), h = f3sub(g, c1);
        F3 p1 = f3add(rot90(q), c1),  g1 = f3add(rot90(h), c1);
        F3 p2 = f3add(f3neg(rot90(f3sub(p1, c1))), c1);
        F3 g2 = f3add(f3neg(rot90(f3sub(g1, c1))), c1);
        F3 p3 = f3add(rot180(q), c1), g3 = f3add(rot180(h), c1);
        F3 p4 = f3add(rotm90(f3sub(p1, c2)), c2);
        F3 g4 = f3add(rotm90(f3sub(g1, c2)), c2);
        F3 p5 = f3add(f3neg(rotm90(f3sub(p3, c2))), c2);
        F3 g5 = f3add(f3neg(rotm90(f3sub(g3, c2))), c2);
        s = projd(p1, g1, f) + projd(p2, g2, f) + projd(p3, g3, f)
          + projd(p4, g4, f) + projd(p5, g5, f);
    }
    #pragma unroll
    for (int m = 16; m >= 1; m >>= 1) s += __shfl_xor(s, m, 32);
    if (lane == 0) lds[wave] = s;
    __syncthreads();
    if (threadIdx.x == 0) {
        float t = 0.f;
        #pragma unroll
        for (int w = 0; w < WAVES_PER_BLOCK; ++w) t += lds[w];
        partials[blockIdx.x] = t;
    }
}

// Single-wave final reduction using V_WMMA_F32_16X16X4_F32 (exact f32 math).
// A carries 64 partials per step (2 per lane), B is all-ones 4x16, so
// D[m][n] = sum_k A[m][k] and C accumulates across iterations. The grand sum
// is sum_m C[m][0] = (per-lane sum of 8 C regs) + shfl_xor(...,16).
__global__ __launch_bounds__(32)
void mvk_wmma_reduce(const float* __restrict__ partials, int n, float inv_count,
                     float* __restrict__ out)
{
    const int lane = threadIdx.x;
    v2f bones = {1.0f, 1.0f};   // all 64 slots of the 4x16 B matrix = 1.0
    v8f c = {};                 // zero accumulator
    const int nfull = n & ~63;
    for (int i = 0; i < nfull; i += 64) {
        v2f a;
        a.x = partials[i + lane];
        a.y = partials[i + 32 + lane];
        // D = A x ones + C  : accumulates sum of all 64 A entries per column
        c = __builtin_amdgcn_wmma_f32_16x16x4_f32(
                /*neg_a=*/false, a, /*neg_b=*/false, bones,
                /*c_mod=*/(short)0, c, /*reuse_a=*/false, /*reuse_b=*/false);
    }
    float t = c[0] + c[1] + c[2] + c[3] + c[4] + c[5] + c[6] + c[7];
    t += __shfl_xor(t, 16, 32);
    // scalar tail for non-multiple-of-64 partial counts (divergence is after WMMA)
    float tail = 0.f;
    for (int i = nfull + lane; i < n; i += 32) tail += partials[i];
    #pragma unroll
    for (int m = 16; m >= 1; m >>= 1) tail += __shfl_xor(tail, m, 32);
    if (lane == 0) {
        float mean = (t + tail) * inv_count;           // sum over 5 views of per-view means
        bool bad = (mean > 1500.0f) || (mean != mean); // >300*5 or NaN -> 0
        out[0] = bad ? 0.0f : mean * 1000.0f;
    }
}

extern "C" void kernel_launch(void* const* d_in, const int* in_sizes, int n_in,
                              void* d_out, int out_size, void* d_ws, size_t ws_size,
                              hipStream_t stream) {
    const float* pred_joints = (const float*)d_in[0];  // [B,J,3]
    const float* pred_trans  = (const float*)d_in[1];  // [B,3]
    const float* gt_joints   = (const float*)d_in[2];  // [B,J,4]
    const float* gt_trans    = (const float*)d_in[3];  // [B,3]
    // d_in[4]=valid, d_in[5]=img_h, d_in[6]=img_w : unused by the reference loss
    const float* focal       = (const float*)d_in[7];  // [B]

    const int B = in_sizes[7];                  // focal_length has B elements
    const int nblocks = B / WAVES_PER_BLOCK;    // B=65536 -> 8192 partials
    float* partials = (float*)d_ws;
    float* out = (float*)d_out;

    mvk_loss_partial<<<nblocks, WAVES_PER_BLOCK * 32, 0, stream>>>(
        pred_joints, pred_trans, gt_joints, gt_trans, focal, partials);

    const float inv_count = 1.0f / ((float)B * (float)J_N * 2.0f);
    mvk_wmma_reduce<<<1, 32, 0, stream>>>(partials, nblocks, inv_count, out);
}